// SAAA_73727408603242
// MI455X (gfx1250) — compile-verified
//
#include <hip/hip_runtime.h>
#include <hip/hip_bf16.h>

typedef float v2f __attribute__((ext_vector_type(2)));
typedef float v8f __attribute__((ext_vector_type(8)));

#define B_  8
#define C_  512
#define HW_ 16384
#define K_  64

static __device__ inline v8f wmma_f32_k4(v2f a, v2f b, v8f c) {
  // D = A(16x4,f32) * B(4x16,f32) + C(16x16,f32)
  return __builtin_amdgcn_wmma_f32_16x16x4_f32(
      /*neg_a=*/false, a, /*neg_b=*/false, b,
      /*c_mod=*/(short)0, c, /*reuse_a=*/false, /*reuse_b=*/false);
}

// ---------------------------------------------------------------------------
// k1: WD[b] = conv_w(512x512) @ descriptors[b](512x64), f32 WMMA, K-loop of 4
// 1024 waves total (8 batches * 32 Mtiles * 4 Ntiles)
// ---------------------------------------------------------------------------
__global__ __launch_bounds__(256) void k1_wd(const float* __restrict__ W,
                                             const float* __restrict__ D,
                                             float* __restrict__ WD) {
  const int lane = threadIdx.x & 31;
  const int wave = threadIdx.x >> 5;
  const int wid  = blockIdx.x * 8 + wave;   // 0..1023
  const int bi   = wid >> 7;
  const int rem  = wid & 127;
  const int m0   = (rem >> 2) * 16;
  const int n0   = (rem & 3) * 16;
  const int off  = (lane >> 4) << 1;        // 0 or 2 (K sub-offset)
  const int ncl  = lane & 15;
  const int row  = m0 + ncl;
  const float* Db = D + bi * (C_ * K_);

  v8f acc = {};
  for (int kk = 0; kk < C_; kk += 4) {
    v2f a = *(const v2f*)(W + row * C_ + kk + off);
    v2f b;
    b.x = Db[(kk + off) * K_ + n0 + ncl];
    b.y = Db[(kk + off + 1) * K_ + n0 + ncl];
    acc = wmma_f32_k4(a, b, acc);
  }
  float* out = WD + bi * (C_ * K_);
  const int rbase = m0 + ((lane >> 4) << 3);
#pragma unroll
  for (int r = 0; r < 8; ++r) out[(rbase + r) * K_ + n0 + ncl] = acc[r];
}

// ---------------------------------------------------------------------------
// k2: per-pixel top-64 of 512 channels + softmax -> atten[b,64,HW]
// One block = 16 pixels. LDS-staged transpose (stride-17 pad, conflict-free),
// full cross-lane bitonic sort of 512 (16 vals/lane, wave32).
// ---------------------------------------------------------------------------
__global__ __launch_bounds__(256) void k2_topk(const float* __restrict__ X,
                                               float* __restrict__ A) {
  __shared__ float lds[C_ * 17];   // [512][16] padded
  __shared__ float att[K_ * 17];   // [64][16] padded
  const int t    = threadIdx.x;
  const int lane = t & 31;
  const int wave = t >> 5;
  const int bi   = blockIdx.x >> 10;         // 1024 tiles of 16 px per batch
  const int n0   = (blockIdx.x & 1023) * 16;
  const float* Xb = X + (size_t)bi * C_ * HW_;

  // coalesced load of 512x16 tile
#pragma unroll
  for (int i = 0; i < 32; ++i) {
    int f = t + i * 256;
    int row = f >> 4, col = f & 15;
    lds[row * 17 + col] = Xb[(size_t)row * HW_ + n0 + col];
  }
  __syncthreads();

#pragma unroll 1
  for (int pp = 0; pp < 2; ++pp) {
    const int p = wave * 2 + pp;   // pixel within tile
    float v[16];
#pragma unroll
    for (int s = 0; s < 16; ++s) v[s] = lds[(s * 32 + lane) * 17 + p];

    // bitonic sort, descending, element m = lane*16 + s
#pragma unroll
    for (int K = 2; K <= 512; K <<= 1) {
#pragma unroll
      for (int J = K >> 1; J >= 1; J >>= 1) {
        if (J >= 16) {                       // cross-lane pass (K >= 32 here)
          const int  ldist   = J >> 4;
          const bool lower   = ((lane & ldist) == 0);
          const bool desc    = (((lane << 4) & K) == 0);
          const bool keepMax = (lower == desc);
#pragma unroll
          for (int s = 0; s < 16; ++s) {
            float w  = __shfl_xor(v[s], ldist, 32);
            float mx = fmaxf(v[s], w), mn = fminf(v[s], w);
            v[s] = keepMax ? mx : mn;
          }
        } else {                             // in-lane pass
#pragma unroll
          for (int s = 0; s < 16; ++s) {
            int l = s ^ J;
            if (l > s) {
              bool desc = ((((lane << 4) | s) & K) == 0);
              float a = v[s], b = v[l];
              float mx = fmaxf(a, b), mn = fminf(a, b);
              v[s] = desc ? mx : mn;
              v[l] = desc ? mn : mx;
            }
          }
        }
      }
    }

    // softmax over top-64 (held by lanes 0..3, slots 0..15, sorted desc)
    float vmax = __shfl(v[0], 0, 32);        // global max = element 0
    float e[16];
    float psum = 0.f;
    if (lane < 4) {
#pragma unroll
      for (int s = 0; s < 16; ++s) { e[s] = __expf(v[s] - vmax); psum += e[s]; }
    }
#pragma unroll
    for (int d = 1; d < 32; d <<= 1) psum += __shfl_xor(psum, d, 32);
    float inv = 1.f / psum;
    if (lane < 4) {
#pragma unroll
      for (int s = 0; s < 16; ++s) att[(lane * 16 + s) * 17 + p] = e[s] * inv;
    }
  }
  __syncthreads();

  // coalesced store of atten tile [64][16]
  float* Ab = A + (size_t)bi * K_ * HW_;
#pragma unroll
  for (int i = 0; i < 4; ++i) {
    int f = t + i * 256;
    int k = f >> 4, col = f & 15;
    Ab[(size_t)k * HW_ + n0 + col] = att[k * 17 + col];
  }
}

// ---------------------------------------------------------------------------
// k3: out[b] = WD[b](512x64) @ atten[b](64x16384) + bias, f32 WMMA.
// Wave = 16 rows x 64 cols; A-fragment (K=64) kept in regs, reused 4x.
// ---------------------------------------------------------------------------
__global__ __launch_bounds__(256) void k3_gemm(const float* __restrict__ WD,
                                               const float* __restrict__ A,
                                               const float* __restrict__ bias,
                                               float* __restrict__ out) {
  const int lane = threadIdx.x & 31;
  const int wave = threadIdx.x >> 5;
  const int wid  = blockIdx.x * 8 + wave;   // 0..65535
  const int bi   = wid >> 13;
  const int rem  = wid & 8191;
  const int m0   = (rem >> 8) * 16;         // 32 M tiles
  const int n0   = (rem & 255) * 64;        // 256 N groups of 64
  const int off  = (lane >> 4) << 1;
  const int ncl  = lane & 15;
  const float* WDb = WD + bi * (C_ * K_);
  const float* Ab  = A + (size_t)bi * K_ * HW_;
  float* Ob        = out + (size_t)bi * C_ * HW_;

  v2f a[16];
#pragma unroll
  for (int ti = 0; ti < 16; ++ti)
    a[ti] = *(const v2f*)(WDb + (m0 + ncl) * K_ + ti * 4 + off);

  const int rbase = m0 + ((lane >> 4) << 3);
  float bs[8];
#pragma unroll
  for (int r = 0; r < 8; ++r) bs[r] = bias[rbase + r];

#pragma unroll 1
  for (int sub = 0; sub < 4; ++sub) {
    const int nbase = n0 + sub * 16 + ncl;
    v8f acc = {};
#pragma unroll
    for (int ti = 0; ti < 16; ++ti) {
      v2f bv;
      bv.x = Ab[(size_t)(ti * 4 + off) * HW_ + nbase];
      bv.y = Ab[(size_t)(ti * 4 + off + 1) * HW_ + nbase];
      acc = wmma_f32_k4(a[ti], bv, acc);
    }
#pragma unroll
    for (int r = 0; r < 8; ++r)
      Ob[(size_t)(rbase + r) * HW_ + nbase] = acc[r] + bs[r];
  }
}

// ---------------------------------------------------------------------------
extern "C" void kernel_launch(void* const* d_in, const int* in_sizes, int n_in,
                              void* d_out, int out_size, void* d_ws, size_t ws_size,
                              hipStream_t stream) {
  const float* desc = (const float*)d_in[0];   // [8,512,64]
  const float* x    = (const float*)d_in[1];   // [8,512,128,128]
  const float* w    = (const float*)d_in[2];   // [512,512,1,1]
  const float* bias = (const float*)d_in[3];   // [512]
  float* out = (float*)d_out;

  float* WD = (float*)d_ws;                               // 8*512*64*4 = 1 MB
  float* At = (float*)((char*)d_ws + (size_t)(1 << 20));  // 8*64*16384*4 = 32 MB

  k1_wd  <<<128,  256, 0, stream>>>(w, desc, WD);
  k2_topk<<<8192, 256, 0, stream>>>(x, At);
  k3_gemm<<<8192, 256, 0, stream>>>(WD, At, bias, out);
}